// CustomDotProdXAttn_86827058856386
// MI455X (gfx1250) — compile-verified
//
#include <hip/hip_runtime.h>
#include <hip/hip_bf16.h>

// ---------------------------------------------------------------------------
// Problem constants (match reference)
// ---------------------------------------------------------------------------
static constexpr int Bc  = 4;
static constexpr int Sc  = 2048;
static constexpr int Tc  = 256;
static constexpr int Nc  = 32;
static constexpr int Hc  = 2048;
static constexpr int Vc  = 1024;
static constexpr int Ac  = 2048;
static constexpr int NHc = 8;
static constexpr int HDc = Ac / NHc;   // 256

typedef __attribute__((ext_vector_type(2))) float v2f;
typedef __attribute__((ext_vector_type(8))) float v8f;

// ---------------------------------------------------------------------------
// K0: per-batch stable selection of mask>0 rows (matches stable argsort trick)
// ---------------------------------------------------------------------------
__global__ void sel_kernel(const int* __restrict__ mask,
                           long long* __restrict__ arowQ,
                           float* __restrict__ keepf) {
  int b = blockIdx.x;
  if (threadIdx.x != 0) return;
  int cnt = 0;
  for (int s = 0; s < Sc && cnt < Tc; ++s) {
    int mv = mask[(long long)b * Sc + s];
    if (mv > 0) {
      arowQ[b * Tc + cnt] = (long long)(b * Sc + s) * Hc;
      keepf[b * Tc + cnt] = (mv >= 2) ? 1.0f : 0.0f;
      ++cnt;
    }
  }
  for (int s = 0; s < Sc && cnt < Tc; ++s) {   // pad with mask==0 rows (stable)
    int mv = mask[(long long)b * Sc + s];
    if (mv <= 0) {
      arowQ[b * Tc + cnt] = (long long)(b * Sc + s) * Hc;
      keepf[b * Tc + cnt] = 0.0f;
      ++cnt;
    }
  }
}

// ---------------------------------------------------------------------------
// Generic fp32 WMMA GEMM:
//   C[m,n] = sum_k A[row(m), k] * B[b_base + k*LDBK + n*LDBN]
// Compile-time strides (LDBK/LDBN/LDC) -> address math folds into load/store
// immediates; running pointers advance by constant deltas.
// Block = 256 threads = 8 waves, block tile 64(M) x 256(N).
// Wave tile 32x64 = 2(A-frag) x 4(B-frag) = 8 x v_wmma_f32_16x16x4_f32 per
// K-step, software-pipelined ping-pong (prefetch k+4 while computing k).
// Requires: M % 64 == 0, N % 256 == 0 (gridDim.x==1 gives N==256), K % 8 == 0.
// ---------------------------------------------------------------------------
template <int LDBK, int LDBN, int LDC, bool GELU, bool SCALE, bool GATHER>
__global__ __launch_bounds__(256) void wmma_gemm_f32(
    const float* __restrict__ A,
    const long long* __restrict__ row_ofs,
    long long row_base, long long row_stride,
    const float* __restrict__ B, long long b_base,
    float* __restrict__ C, long long c_base,
    int K,
    const float* __restrict__ row_scale) {
  const int wave = threadIdx.x >> 5;
  const int lane = threadIdx.x & 31;
  const int half = lane >> 4;      // 0: K pair {0,1}, 1: K pair {2,3}
  const int lidx = lane & 15;
  const int wm = wave >> 2;        // 0..1  (32-row granule)
  const int wn = wave & 3;         // 0..3  (64-col granule)
  const int m0 = blockIdx.y * 64 + wm * 32;
  const int n0 = blockIdx.x * 256 + wn * 64;

  // A row pointers (rows replicated across both lane halves; halves own K pairs)
  const float* ap[2];
#pragma unroll
  for (int mi = 0; mi < 2; ++mi) {
    const int mrow = m0 + mi * 16 + lidx;
    const long long ar =
        GATHER ? row_ofs[mrow] : (row_base + (long long)mrow * row_stride);
    ap[mi] = A + ar + 2 * half;
  }
  // B column pointers per N-tile
  const float* bp[4];
#pragma unroll
  for (int j = 0; j < 4; ++j)
    bp[j] = B + b_base + (long long)(n0 + j * 16 + lidx) * LDBN +
            (long long)(2 * half) * LDBK;

  v8f acc[2][4] = {};
  v2f aA[2], bA[4], aB[2], bB[4];

  auto load_set = [&](int ofs, v2f (&aF)[2], v2f (&bF)[4]) {
#pragma unroll
    for (int mi = 0; mi < 2; ++mi)
      aF[mi] = *reinterpret_cast<const v2f*>(ap[mi] + ofs);   // b64
#pragma unroll
    for (int j = 0; j < 4; ++j) {
      bF[j].x = bp[j][ofs * LDBK];
      bF[j].y = bp[j][ofs * LDBK + LDBK];
    }
  };
  auto mma_set = [&](v2f (&aF)[2], v2f (&bF)[4]) {
#pragma unroll
    for (int mi = 0; mi < 2; ++mi)
#pragma unroll
      for (int j = 0; j < 4; ++j)
        acc[mi][j] = __builtin_amdgcn_wmma_f32_16x16x4_f32(
            false, aF[mi], false, bF[j], (short)0, acc[mi][j], false, false);
  };

  load_set(0, aA, bA);
#pragma unroll 1
  for (int kk = 0; kk + 8 < K; kk += 8) {
    load_set(4, aB, bB);   // prefetch k+4 while computing k
    mma_set(aA, bA);
    load_set(8, aA, bA);   // prefetch k+8 while computing k+4
    mma_set(aB, bB);
#pragma unroll
    for (int mi = 0; mi < 2; ++mi) ap[mi] += 8;
#pragma unroll
    for (int j = 0; j < 4; ++j) bp[j] += (long long)8 * LDBK;
  }
  load_set(4, aB, bB);     // last pair (pointers sit at K-8)
  mma_set(aA, bA);
  mma_set(aB, bB);

  // Optional per-row scale (keep mask), hoisted out of store loops
  float rs[2][8];
  if (SCALE) {
#pragma unroll
    for (int mi = 0; mi < 2; ++mi)
#pragma unroll
      for (int r = 0; r < 8; ++r)
        rs[mi][r] = row_scale[m0 + mi * 16 + half * 8 + r];
  }

  // C 16x16 f32 layout: VGPR r -> (m = tile_m + 8*half + r, n = tile_n + lidx)
#pragma unroll
  for (int mi = 0; mi < 2; ++mi)
#pragma unroll
    for (int j = 0; j < 4; ++j) {
      const int n = n0 + j * 16 + lidx;
#pragma unroll
      for (int r = 0; r < 8; ++r) {
        const int m = m0 + mi * 16 + half * 8 + r;
        float val = acc[mi][j][r];
        if (GELU) val = 0.5f * val * (1.0f + erff(val * 0.70710678118654752f));
        if (SCALE) val *= rs[mi][r];
        C[c_base + (long long)m * LDC + n] = val;
      }
    }
}

// ---------------------------------------------------------------------------
// K3: LayerNorm over last dim (V=1024); one wave per row, float4 + shfl reduce
// ---------------------------------------------------------------------------
__global__ __launch_bounds__(256) void ln_kernel(const float* __restrict__ feat,
                                                 const float* __restrict__ gamma,
                                                 const float* __restrict__ beta,
                                                 float* __restrict__ xn,
                                                 int rows) {
  const int row = blockIdx.x * 8 + (threadIdx.x >> 5);
  const int lane = threadIdx.x & 31;
  if (row >= rows) return;
  const float* __restrict__ x = feat + (long long)row * Vc;
  const float4* __restrict__ x4 = reinterpret_cast<const float4*>(x);
  float s = 0.f, s2 = 0.f;
  for (int i = lane; i < Vc / 4; i += 32) {
    float4 t = x4[i];
    s  += t.x + t.y + t.z + t.w;
    s2 += t.x * t.x + t.y * t.y + t.z * t.z + t.w * t.w;
  }
#pragma unroll
  for (int o = 16; o; o >>= 1) {
    s  += __shfl_xor(s,  o, 32);
    s2 += __shfl_xor(s2, o, 32);
  }
  const float mu  = s * (1.0f / Vc);
  const float var = s2 * (1.0f / Vc) - mu * mu;
  const float inv = rsqrtf(var + 1e-5f);
  const float4* __restrict__ g4 = reinterpret_cast<const float4*>(gamma);
  const float4* __restrict__ b4 = reinterpret_cast<const float4*>(beta);
  float4* __restrict__ y4 = reinterpret_cast<float4*>(xn + (long long)row * Vc);
  for (int i = lane; i < Vc / 4; i += 32) {
    float4 t = x4[i], gg = g4[i], bb = b4[i], o;
    o.x = (t.x - mu) * inv * gg.x + bb.x;
    o.y = (t.y - mu) * inv * gg.y + bb.y;
    o.z = (t.z - mu) * inv * gg.z + bb.z;
    o.w = (t.w - mu) * inv * gg.w + bb.w;
    y4[i] = o;
  }
}

// ---------------------------------------------------------------------------
// K5: fused attention. 1 block per (b,t); wave = head, lane = vision token.
//   logits[h,n] = (1/sqrt(A)) * dot(feat[b,t,n,:], qk[b,t,h,:])   (qk in LDS)
//   softmax over 32 lanes (wave32 shuffles)  -> writes `attn` output
//   gbar[h,w]   = sum_n attn[h,n] * g[b,t,n,w]
// ---------------------------------------------------------------------------
__global__ __launch_bounds__(256) void attn_kernel(
    const float* __restrict__ feat,
    const float* __restrict__ qk,
    const float* __restrict__ g,
    float* __restrict__ attn_out,
    float* __restrict__ gbar) {
  const int bt = blockIdx.x;
  __shared__ __align__(16) float s_qk[NHc * Vc];  // 32 KB
  __shared__ float s_attn[NHc][Nc];

  const float4* __restrict__ qk4 =
      reinterpret_cast<const float4*>(qk + (long long)bt * NHc * Vc);
  float4* __restrict__ sq4 = reinterpret_cast<float4*>(s_qk);
  for (int i = threadIdx.x; i < NHc * Vc / 4; i += 256) sq4[i] = qk4[i];
  __syncthreads();

  const int h = threadIdx.x >> 5;     // head
  const int lane = threadIdx.x & 31;  // vision token n

  const float4* __restrict__ f4 = reinterpret_cast<const float4*>(
      feat + ((long long)bt * Nc + lane) * Vc);
  const float4* __restrict__ q4h =
      reinterpret_cast<const float4*>(&s_qk[h * Vc]);
  float acc = 0.f;
  for (int i = 0; i < Vc / 4; ++i) {
    float4 fv = f4[i], qv = q4h[i];
    acc += fv.x * qv.x + fv.y * qv.y + fv.z * qv.z + fv.w * qv.w;
  }
  const float logit = acc * 0.02209708691207961f;  // 1/sqrt(2048)

  float mx = logit;
#pragma unroll
  for (int o = 16; o; o >>= 1) mx = fmaxf(mx, __shfl_xor(mx, o, 32));
  float e = __expf(logit - mx);
  float sum = e;
#pragma unroll
  for (int o = 16; o; o >>= 1) sum += __shfl_xor(sum, o, 32);
  const float a = e / sum;

  attn_out[(long long)bt * NHc * Nc + h * Nc + lane] = a;
  s_attn[h][lane] = a;
  __syncthreads();

  // gbar[h,w] = sum_n attn[h,n] * g[n,w]; lanes stride over w (coalesced)
  const float* __restrict__ grow = g + (long long)bt * Nc * Vc;
  float* __restrict__ gb = gbar + (long long)bt * NHc * Vc + (long long)h * Vc;
  for (int w = lane; w < Vc; w += 32) {
    float sacc = 0.f;
#pragma unroll 8
    for (int n = 0; n < Nc; ++n) sacc += s_attn[h][n] * grow[n * Vc + w];
    gb[w] = sacc;
  }
}

// ---------------------------------------------------------------------------
// Host launcher
// ---------------------------------------------------------------------------
extern "C" void kernel_launch(void* const* d_in, const int* in_sizes, int n_in,
                              void* d_out, int out_size, void* d_ws, size_t ws_size,
                              hipStream_t stream) {
  const float* hidden = (const float*)d_in[0];
  const float* feats  = (const float*)d_in[1];
  const int*   mask   = (const int*)d_in[2];
  const float* Wq     = (const float*)d_in[3];
  const float* Wk     = (const float*)d_in[4];
  const float* gamma  = (const float*)d_in[5];
  const float* beta   = (const float*)d_in[6];
  const float* Wv1    = (const float*)d_in[7];
  const float* Wv2    = (const float*)d_in[8];
  const float* Wout   = (const float*)d_in[9];

  float* out      = (float*)d_out;                      // (B,T,H)
  float* attn_out = out + (size_t)Bc * Tc * Hc;         // (B,T,8,1,32)

  // Workspace layout
  char* w = (char*)d_ws;
  size_t off = 0;
  auto alloc = [&](size_t bytes) -> void* {
    void* p = w + off;
    off = (off + bytes + 255) & ~(size_t)255;
    return p;
  };
  const int BT  = Bc * Tc;         // 1024
  const int BTN = Bc * Tc * Nc;    // 32768
  float*     q_buf  = (float*)alloc((size_t)BT * Ac * 4);        //   8 MB
  float*     qk_buf = (float*)alloc((size_t)BT * NHc * Vc * 4);  //  32 MB
  float*     xn_buf = (float*)alloc((size_t)BTN * Vc * 4);       // 128 MB
  float*     g_buf  = (float*)alloc((size_t)BTN * Vc * 4);       // 128 MB
  float*     gbar   = (float*)alloc((size_t)BT * NHc * Vc * 4);  //  32 MB
  float*     ctx    = (float*)alloc((size_t)BT * Ac * 4);        //   8 MB
  long long* arowQ  = (long long*)alloc((size_t)BT * 8);
  float*     keepf  = (float*)alloc((size_t)BT * 4);

  // K0: row selection + keep mask
  sel_kernel<<<Bc, 32, 0, stream>>>(mask, arowQ, keepf);

  // K1: q = gather(hidden)[sel] @ Wq       M=1024 N=2048 K=2048
  wmma_gemm_f32<Ac, 1, Ac, false, false, true>
      <<<dim3(Ac / 256, BT / 64), 256, 0, stream>>>(
          hidden, arowQ, 0, 0, Wq, 0, q_buf, 0, Hc, nullptr);

  // K2: qk[bt,h,v] = sum_j q[bt,h*256+j] * Wk[v,h*256+j]  (8 head GEMMs)
  //     B is a transposed read of Wk: ldbk=1, ldbn=A. M=1024 N=1024 K=256
  for (int h = 0; h < NHc; ++h) {
    wmma_gemm_f32<1, Ac, NHc * Vc, false, false, false>
        <<<dim3(Vc / 256, BT / 64), 256, 0, stream>>>(
            q_buf, nullptr, (long long)h * HDc, Ac,
            Wk, (long long)h * HDc,
            qk_buf, (long long)h * Vc, HDc, nullptr);
  }

  // K3: LayerNorm of vision feats
  ln_kernel<<<BTN / 8, 256, 0, stream>>>(feats, gamma, beta, xn_buf, BTN);

  // K4: g = gelu(xn @ Wv1)                 M=32768 N=1024 K=1024
  wmma_gemm_f32<Vc, 1, Vc, true, false, false>
      <<<dim3(Vc / 256, BTN / 64), 256, 0, stream>>>(
          xn_buf, nullptr, 0, Vc, Wv1, 0, g_buf, 0, Vc, nullptr);

  // K5: fused logits+softmax (writes attn output) + gbar
  attn_kernel<<<BT, 256, 0, stream>>>(feats, qk_buf, g_buf, attn_out, gbar);

  // K6: ctx[bt,h*256+j] = sum_w gbar[bt,h,w] * Wv2[w,h*256+j]  (8 head GEMMs)
  //     M=1024 N=256 K=1024
  for (int h = 0; h < NHc; ++h) {
    wmma_gemm_f32<Ac, 1, Ac, false, false, false>
        <<<dim3(1, BT / 64), 256, 0, stream>>>(
            gbar, nullptr, (long long)h * Vc, (long long)NHc * Vc,
            Wv2, (long long)h * HDc,
            ctx, (long long)h * HDc, Vc, nullptr);
  }

  // K7: out = (ctx @ Wout) * keep[row]     M=1024 N=2048 K=2048
  wmma_gemm_f32<Hc, 1, Hc, false, true, false>
      <<<dim3(Hc / 256, BT / 64), 256, 0, stream>>>(
          ctx, nullptr, 0, Ac, Wout, 0, out, 0, Ac, keepf);

  (void)in_sizes; (void)n_in; (void)out_size; (void)ws_size;
}